// PolyAttn_6880537608224
// MI455X (gfx1250) — compile-verified
//
#include <hip/hip_runtime.h>

// ---------------------------------------------------------------------------
// PolyAttn on MI455X (gfx1250, wave32).
//
// Reference math collapses: a = (alpha*QK^T+1)^4 / |...| == all-ones matrix,
// so  o[b,n,:] = ((sum_m x[b,m,:]) @ w_v) @ w_o   for every n.
// => bandwidth-bound (~64MB traffic, ~3us @ 23.3 TB/s). The two surviving
// GEMMs are [16x1024]x[1024x1024]; done in full fp32 via V_WMMA_F32_16X16X4_F32.
// ---------------------------------------------------------------------------

typedef __attribute__((ext_vector_type(2))) float v2f;
typedef __attribute__((ext_vector_type(8))) float v8f;

#define B_ 4
#define N_ 2048
#define D_ 1024
#define H_ 1024

// ---- zero a small fp32 region (ws is poisoned with 0xAA by the harness) ----
__global__ void zero_f32(float* __restrict__ p, int n) {
    int i = blockIdx.x * blockDim.x + threadIdx.x;
    if (i < n) p[i] = 0.0f;
}

// ---- xsum[b,d] = sum_n x[b,n,d]  (streams all 32MB of x, coalesced) --------
__global__ void colsum_kernel(const float* __restrict__ x,
                              float* __restrict__ xsum) {
    const int d  = blockIdx.x * 256 + threadIdx.x;   // 0..D-1, coalesced
    const int n0 = blockIdx.y * 256;                 // n-slice
    const int b  = blockIdx.z;
    const float* p = x + ((size_t)b * N_ + n0) * D_ + d;
    float s = 0.0f;
#pragma unroll 8
    for (int i = 0; i < 256; ++i) s += p[(size_t)i * D_];
    atomicAdd(&xsum[b * D_ + d], s);                 // global_atomic_add_f32
}

// ---- Dst[0:16, n0:n0+16] = A[16xK] * Bm[Kx16]  via V_WMMA_F32_16X16X4_F32 --
// One wave per 16-wide output tile. fp32 A-frag layout (ISA 7.12.2):
//   lanes 0-15 : M=lane,    VGPR0=K+0, VGPR1=K+1
//   lanes 16-31: M=lane-16, VGPR0=K+2, VGPR1=K+3
// B/C/D: one row striped across 16 lanes per VGPR (low/high lane halves map
// to K/M offset +2 / +8 respectively).
__global__ void __launch_bounds__(32)
gemm16_f32wmma(const float* __restrict__ A, int lda,
               const float* __restrict__ Bm, int ldb,
               float* __restrict__ Dst, int ldd, int K) {
    const int lane = threadIdx.x;       // 0..31, EXEC all-ones (no divergence)
    const int n0   = blockIdx.x * 16;
    const int half = lane >> 4;         // 0 | 1
    const int mr   = lane & 15;         // A-row / B,D-col within tile

    v8f c = {};
    for (int k = 0; k < K; k += 4) {
        const int kb = k + half * 2;
        v2f a, b;
        a.x = A[mr * lda + kb];
        a.y = A[mr * lda + kb + 1];
        b.x = Bm[(size_t)kb       * ldb + n0 + mr];
        b.y = Bm[(size_t)(kb + 1) * ldb + n0 + mr];
        // (neg_a, A, neg_b, B, c_mod, C, reuse_a, reuse_b)
        c = __builtin_amdgcn_wmma_f32_16x16x4_f32(false, a, false, b,
                                                  (short)0, c, false, false);
    }

    const int mbase = half * 8;
#pragma unroll
    for (int j = 0; j < 8; ++j)
        Dst[(mbase + j) * ldd + n0 + mr] = c[j];
}

// ---- out[b,n,:] = rowv[b,:]  (32MB of b128 stores) -------------------------
__global__ void __launch_bounds__(256)
bcast_kernel(const float* __restrict__ rowv, float* __restrict__ out) {
    const int bn = blockIdx.x;          // 0 .. B*N-1
    const int b  = bn >> 11;            // N_ == 2048
    const float4 v = ((const float4*)(rowv + b * D_))[threadIdx.x];
    ((float4*)(out + (size_t)bn * D_))[threadIdx.x] = v;   // global_store_b128
}

extern "C" void kernel_launch(void* const* d_in, const int* in_sizes, int n_in,
                              void* d_out, int out_size, void* d_ws, size_t ws_size,
                              hipStream_t stream) {
    const float* x     = (const float*)d_in[0];   // [B, N, D]
    const float* w_qkv = (const float*)d_in[1];   // [D, 3H] row-major
    const float* w_o   = (const float*)d_in[2];   // [H, D]  row-major
    // d_in[3] = alpha: provably irrelevant (a == 1 for any alpha).
    float* out = (float*)d_out;

    // Workspace: three 16-row padded fp32 panels (rows 4..15 stay zero).
    float* xsum = (float*)d_ws;           // [16, D_]
    float* vsum = xsum + 16 * D_;         // [16, H_]
    float* rowv = vsum + 16 * H_;         // [16, D_]

    // 1) xsum = 0 (atomic accumulation target; also zero-pads rows 4..15)
    zero_f32<<<(16 * D_) / 256, 256, 0, stream>>>(xsum, 16 * D_);

    // 2) xsum[b,:] = sum_n x[b,n,:]
    colsum_kernel<<<dim3(D_ / 256, N_ / 256, B_), 256, 0, stream>>>(x, xsum);

    // 3) vsum = xsum @ w_v   (w_v = w_qkv[:, 2H:3H], row stride 3H)
    gemm16_f32wmma<<<H_ / 16, 32, 0, stream>>>(xsum, D_,
                                               w_qkv + 2 * H_, 3 * H_,
                                               vsum, H_, D_);

    // 4) rowv = vsum @ w_o
    gemm16_f32wmma<<<D_ / 16, 32, 0, stream>>>(vsum, H_,
                                               w_o, D_,
                                               rowv, D_, H_);

    // 5) out[b,n,:] = rowv[b,:]
    bcast_kernel<<<B_ * N_, 256, 0, stream>>>(rowv, out);
}